// GraphMultiheadAttention_14663018348779
// MI455X (gfx1250) — compile-verified
//
#include <hip/hip_runtime.h>

// GraphMultiheadAttention for MI455X (gfx1250):
//  - fp32 math via V_WMMA_F32_16X16X4_F32
//  - LDS tile staging via GLOBAL_LOAD_ASYNC_TO_LDS_B128 (ASYNCcnt-tracked)
//  - double-buffered K-loop in the GEMMs (copy/compute overlap)

#define BDIM 16
#define NDIM 512
#define EDIM 768
#define HDIM 12
#define DDIM 64

typedef __attribute__((ext_vector_type(2))) float v2f;
typedef __attribute__((ext_vector_type(8))) float v8f;

static __device__ __forceinline__ v8f wmma_f32(v2f a, v2f b, v8f c) {
  // D = A(16x4) * B(4x16) + C(16x16), fp32, wave32
  return __builtin_amdgcn_wmma_f32_16x16x4_f32(
      /*neg_a=*/false, a, /*neg_b=*/false, b,
      /*c_mod=*/(short)0, c, /*reuse_a=*/false, /*reuse_b=*/false);
}

// 16-byte async copy: global memory -> LDS, tracked by ASYNCcnt.
static __device__ __forceinline__ void async_copy16(const float* g, float* l) {
  const unsigned lds = (unsigned)(uintptr_t)l;   // low 32 bits = LDS offset
  const unsigned long long ga = (unsigned long long)(uintptr_t)g;
  asm volatile("global_load_async_to_lds_b128 %0, %1, off"
               :
               : "v"(lds), "v"(ga)
               : "memory");
}

static __device__ __forceinline__ void wait_async0() {
#if __has_builtin(__builtin_amdgcn_s_wait_asynccnt)
  __builtin_amdgcn_s_wait_asynccnt(0);
#else
  asm volatile("s_wait_asynccnt 0" ::: "memory");
#endif
}

// ---------------------------------------------------------------------------
// GEMM: Y = X[M=8192,K=768] @ W[768,768]^T + bias, 64x64 tile per 256-thr
// block, double-buffered 16-wide K chunks staged with async-to-LDS copies.
// mode 0: Y scattered to [B,H,N,D] (head == blockIdx.y since N-tile == 64)
// mode 1: Y row-major [8192 x 768]
// ---------------------------------------------------------------------------
__global__ __launch_bounds__(256) void gemm_bias_kernel(
    const float* __restrict__ X, const float* __restrict__ W,
    const float* __restrict__ bias, float* __restrict__ Y, int mode) {
  __shared__ __align__(16) float As[2][64 * 20];  // [m][i], stride 20
  __shared__ __align__(16) float Bs[2][64 * 20];  // [o][i], stride 20

  const int t = threadIdx.x;
  const int w = t >> 5, l = t & 31;
  const int lm = l & 15, lh = l >> 4;
  const int gm0 = blockIdx.x * 64;
  const int ob = blockIdx.y * 64;
  const int mrow = (w & 3) * 16;   // wave's row tile within 64
  const int ccol = (w >> 2) * 32;  // wave's pair of col tiles

  // one b128 per thread per matrix per chunk: 256 * 4 = 1024 floats
  const int sr = t >> 2;          // 0..63
  const int si = (t & 3) * 4;     // 0,4,8,12
  const float* Xrow = X + (size_t)(gm0 + sr) * EDIM + si;
  const float* Wrow = W + (size_t)(ob + sr) * EDIM + si;
  float* asl = &As[0][sr * 20 + si];
  float* bsl = &Bs[0][sr * 20 + si];
  const int bufstride = 64 * 20;

  v8f acc0 = {};
  v8f acc1 = {};

  // prologue: stage chunk 0 into buffer 0
  async_copy16(Xrow, asl);
  async_copy16(Wrow, bsl);

  for (int kt = 0; kt < 48; ++kt) {
    const int p = kt & 1;
    wait_async0();
    __syncthreads();
    if (kt + 1 < 48) {  // prefetch next chunk into the other buffer
      const int ib = (kt + 1) * 16;
      async_copy16(Xrow + ib, asl + (p ^ 1) * bufstride - p * 0 +
                                  ((p ^ 1) - 0) * 0);  // see below
      async_copy16(Wrow + ib, bsl + (p ^ 1) * bufstride);
    }
    const float* Ap = As[p];
    const float* Bp = Bs[p];
#pragma unroll
    for (int ks = 0; ks < 4; ++ks) {
      const int kk = ks * 4 + lh * 2;
      const v2f a = *(const v2f*)&Ap[(mrow + lm) * 20 + kk];
      const v2f b0 = *(const v2f*)&Bp[(ccol + lm) * 20 + kk];
      const v2f b1 = *(const v2f*)&Bp[(ccol + 16 + lm) * 20 + kk];
      acc0 = wmma_f32(a, b0, acc0);
      acc1 = wmma_f32(a, b1, acc1);
    }
  }

  const int bidx = gm0 / NDIM;  // 64-row tile lies within one batch
  const int h = blockIdx.y;     // 64-col tile == one head (mode 0)
#pragma unroll
  for (int tile = 0; tile < 2; ++tile) {
    v8f acc = tile ? acc1 : acc0;
    const int go = ob + ccol + tile * 16 + lm;
    const float bb = bias[go];
#pragma unroll
    for (int r = 0; r < 8; ++r) {
      const int gm = gm0 + mrow + r + lh * 8;
      const float val = acc[r] + bb;
      if (mode == 0) {
        const int n = gm - bidx * NDIM;
        const int d = go - h * DDIM;
        Y[(((size_t)bidx * HDIM + h) * NDIM + n) * DDIM + d] = val;
      } else {
        Y[(size_t)gm * EDIM + go] = val;
      }
    }
  }
}

// ---------------------------------------------------------------------------
// Attention: one 128-thread block (4 waves) per 16 query rows of one (b,h).
// Phase 1: S = Q K^T * scale + spatial + edge  (WMMA, K async-staged in LDS)
// Phase 2: row softmax (shfl reductions), writes attn_weights
// Phase 3: O = P V  (WMMA, V async-staged in LDS), writes [B,N,E]
// ---------------------------------------------------------------------------
__global__ __launch_bounds__(128) void attn_kernel(
    const float* __restrict__ Q, const float* __restrict__ K,
    const float* __restrict__ V, const float* __restrict__ SP,
    const float* __restrict__ EB, float* __restrict__ O,
    float* __restrict__ AW) {
  __shared__ __align__(16) float sS[16 * 514];  // score/prob rows, stride 514
  __shared__ __align__(16) float Ts[64 * 68];   // K/V tile [key][d], stride 68

  const int t = threadIdx.x;
  const int w = t >> 5, l = t & 31;
  const int lm = l & 15, lh = l >> 4;
  const int gm0 = blockIdx.x * 16;
  const int h = blockIdx.y, b = blockIdx.z;
  const int bh = b * HDIM + h;
  const float scale = 0.125f;  // 1/sqrt(64)

  const float* Qb = Q + ((size_t)bh * NDIM + gm0) * DDIM;
  const float* Kb = K + (size_t)bh * NDIM * DDIM;
  const float* Vb = V + (size_t)bh * NDIM * DDIM;

  // Q-block A fragments (16x64), loaded once, reused for all 32 col tiles.
  v2f af[16];
#pragma unroll
  for (int ks = 0; ks < 16; ++ks) {
    const int kk = ks * 4 + lh * 2;
    af[ks] = *(const v2f*)&Qb[lm * DDIM + kk];
  }

  // ---- Phase 1: scores ----
  for (int cb = 0; cb < 8; ++cb) {  // 64 keys per chunk
#pragma unroll
    for (int p = 0; p < 8; ++p) {   // 8 x b128 per thread = 4096 floats
      const int idx = (p * 128 + t) * 4;
      async_copy16(Kb + (size_t)cb * 4096 + idx,
                   &Ts[(idx >> 6) * 68 + (idx & 63)]);
    }
    wait_async0();
    __syncthreads();
    v8f acc = {};
#pragma unroll
    for (int ks = 0; ks < 16; ++ks) {
      const int kk = ks * 4 + lh * 2;
      const v2f bf = *(const v2f*)&Ts[(w * 16 + lm) * 68 + kk];
      acc = wmma_f32(af[ks], bf, acc);
    }
    const size_t biasrow = ((size_t)b * NDIM + gm0) * NDIM;
    const int jg = cb * 64 + w * 16 + lm;
#pragma unroll
    for (int r = 0; r < 8; ++r) {
      const int m = r + lh * 8;
      const float sv = acc[r] * scale + SP[biasrow + (size_t)m * NDIM + jg] +
                       EB[biasrow + (size_t)m * NDIM + jg];
      sS[m * 514 + jg] = sv;
    }
    __syncthreads();
  }

  // ---- Phase 2: softmax (4 rows per wave) ----
#pragma unroll
  for (int rr = 0; rr < 4; ++rr) {
    const int m = w * 4 + rr;
    float mx = -3.402823466e38f;
    for (int c = l; c < NDIM; c += 32) mx = fmaxf(mx, sS[m * 514 + c]);
#pragma unroll
    for (int off = 16; off; off >>= 1) mx = fmaxf(mx, __shfl_xor(mx, off, 32));
    float sum = 0.f;
    for (int c = l; c < NDIM; c += 32) {
      const float e = __expf(sS[m * 514 + c] - mx);
      sS[m * 514 + c] = e;
      sum += e;
    }
#pragma unroll
    for (int off = 16; off; off >>= 1) sum += __shfl_xor(sum, off, 32);
    const float inv = 1.f / sum;
    float* awrow = AW + ((size_t)bh * NDIM + gm0 + m) * NDIM;
    for (int c = l; c < NDIM; c += 32) {
      const float p = sS[m * 514 + c] * inv;
      sS[m * 514 + c] = p;
      awrow[c] = p;
    }
  }
  __syncthreads();

  // ---- Phase 3: O = P @ V (16x64), wave w owns col tile w ----
  v8f oacc = {};
  for (int kc = 0; kc < 8; ++kc) {  // 64 keys per chunk
#pragma unroll
    for (int p = 0; p < 8; ++p) {
      const int idx = (p * 128 + t) * 4;
      async_copy16(Vb + (size_t)kc * 4096 + idx,
                   &Ts[(idx >> 6) * 68 + (idx & 63)]);
    }
    wait_async0();
    __syncthreads();
#pragma unroll
    for (int ks = 0; ks < 16; ++ks) {
      const int kk = ks * 4 + lh * 2;
      const int kg = kc * 64 + kk;
      const v2f a2 = *(const v2f*)&sS[lm * 514 + kg];
      v2f b2;
      b2.x = Ts[kk * 68 + w * 16 + lm];        // B[k][n] = V[key k][d]
      b2.y = Ts[(kk + 1) * 68 + w * 16 + lm];
      oacc = wmma_f32(a2, b2, oacc);
    }
    __syncthreads();
  }
#pragma unroll
  for (int r = 0; r < 8; ++r) {
    const int n = gm0 + r + lh * 8;
    const int e = h * DDIM + w * 16 + lm;
    O[((size_t)b * NDIM + n) * EDIM + e] = oacc[r];
  }
}

// ---------------------------------------------------------------------------
extern "C" void kernel_launch(void* const* d_in, const int* in_sizes, int n_in,
                              void* d_out, int out_size, void* d_ws,
                              size_t ws_size, hipStream_t stream) {
  (void)in_sizes; (void)n_in; (void)out_size; (void)ws_size;
  const float* query = (const float*)d_in[0];
  const float* key_in = (const float*)d_in[1];
  const float* value = (const float*)d_in[2];
  const float* sp = (const float*)d_in[3];
  const float* eb = (const float*)d_in[4];
  const float* Wq = (const float*)d_in[5];
  const float* bq = (const float*)d_in[6];
  const float* Wk = (const float*)d_in[7];
  const float* bk = (const float*)d_in[8];
  const float* Wv = (const float*)d_in[9];
  const float* bv = (const float*)d_in[10];
  const float* Wo = (const float*)d_in[11];
  const float* bo = (const float*)d_in[12];

  float* out = (float*)d_out;
  float* attn_out = out;                             // [B,N,E]
  float* attn_w = out + (size_t)BDIM * NDIM * EDIM;  // [B,H,N,N]

  float* ws = (float*)d_ws;
  const size_t qkv = (size_t)BDIM * HDIM * NDIM * DDIM;  // 6,291,456
  float* Qw = ws;
  float* Kw = ws + qkv;
  float* Vw = ws + 2 * qkv;
  float* Ow = ws + 3 * qkv;  // [B,N,E]

  const dim3 gg(128, 12, 1), bg(256);
  gemm_bias_kernel<<<gg, bg, 0, stream>>>(query, Wq, bq, Qw, 0);
  gemm_bias_kernel<<<gg, bg, 0, stream>>>(key_in, Wk, bk, Kw, 0);
  gemm_bias_kernel<<<gg, bg, 0, stream>>>(value, Wv, bv, Vw, 0);

  attn_kernel<<<dim3(32, 12, 16), dim3(128), 0, stream>>>(Qw, Kw, Vw, sp, eb,
                                                          Ow, attn_w);

  gemm_bias_kernel<<<gg, bg, 0, stream>>>(Ow, Wo, bo, attn_out, 1);
}